// SelfAttention_13477607374926
// MI455X (gfx1250) — compile-verified
//
#include <hip/hip_runtime.h>

typedef float v2f __attribute__((ext_vector_type(2)));
typedef float v8f __attribute__((ext_vector_type(8)));

#define WMMA_F32(a, b, c) \
    __builtin_amdgcn_wmma_f32_16x16x4_f32(false, (a), false, (b), (short)0, (c), false, false)

#define EMBED 1024
#define THREE_EMBED 3072
#define NHEAD 16
#define HD 64
#define SEQ 2048
#define BATCH 4

// ---------------------------------------------------------------------------
// Kernel 1: QKV projection GEMM.  qkv = x @ W + b
// x: [8192, 1024], W: [1024, 3072], b: [3072]
// One wave computes a 32 (M) x 64 (N) tile: two A fragments share each B
// fragment -> 8 WMMAs per 10 loads per k-step of 4.
// Waves within a block share the N-tile (B-fragment addresses identical
// across the 8 waves -> L0 reuse) and take consecutive M-tiles.
// Output scattered head-major: q/k/v each [B, H, T, hd] in workspace.
// ---------------------------------------------------------------------------
__global__ __launch_bounds__(256) void qkv_gemm_kernel(
    const float* __restrict__ x, const float* __restrict__ w,
    const float* __restrict__ bias,
    float* __restrict__ qb, float* __restrict__ kb, float* __restrict__ vb)
{
    const int lane   = threadIdx.x & 31;
    const int wave   = threadIdx.x >> 5;
    const int lane16 = lane & 15;
    const int laneHi = lane >> 4;

    const int wid   = blockIdx.x * 8 + wave;     // 12288 waves total
    const int mtile = wid & 255;                 // 0..255  (8192/32)
    const int ntile = wid >> 8;                  // 0..47   (3072/64)
    const int m0    = mtile * 32;
    const int n0    = ntile * 64;

    const float* arow0 = x + (long)(m0 + lane16) * EMBED + 2 * laneHi;
    const float* arow1 = arow0 + 16 * EMBED;

    v8f acc0[4] = {};
    v8f acc1[4] = {};

    #pragma unroll 4
    for (int k0 = 0; k0 < EMBED; k0 += 4) {
        v2f a0 = *(const v2f*)(arow0 + k0);
        v2f a1 = *(const v2f*)(arow1 + k0);
        const int kk = k0 + 2 * laneHi;
        const float* wp = w + (long)kk * THREE_EMBED + n0 + lane16;
        #pragma unroll
        for (int c = 0; c < 4; ++c) {
            v2f b;
            b.x = wp[c * 16];                 // W[kk  ][n0 + 16c + lane16]
            b.y = wp[c * 16 + THREE_EMBED];   // W[kk+1][n0 + 16c + lane16]
            acc0[c] = WMMA_F32(a0, b, acc0[c]);
            acc1[c] = WMMA_F32(a1, b, acc1[c]);
        }
    }

    // Epilogue: add bias, scatter to q/k/v in [B, H, T, hd] layout.
    const int section = n0 / EMBED;             // 0=q 1=k 2=v
    const int h       = (n0 % EMBED) / HD;
    float* dst = (section == 0) ? qb : (section == 1) ? kb : vb;

    #pragma unroll
    for (int c = 0; c < 4; ++c) {
        const float bv = bias[n0 + c * 16 + lane16];
        const int d = c * 16 + lane16;
        #pragma unroll
        for (int g = 0; g < 2; ++g) {
            #pragma unroll
            for (int vv = 0; vv < 8; ++vv) {
                const int m  = m0 + 16 * g + vv + 8 * laneHi;
                const int bg = m >> 11;              // batch
                const int t  = m & 2047;             // time
                const float val = (g == 0 ? acc0[c][vv] : acc1[c][vv]) + bv;
                dst[(((long)(bg * NHEAD + h)) * SEQ + t) * HD + d] = val;
            }
        }
    }
}

// ---------------------------------------------------------------------------
// Kernel 2: causal flash attention.  One wave per (b, h, 32-query tile).
// Two 16-query halves share every K and V fragment: per 16-key tile,
// 16 K-loads feed 32 S-WMMAs and 32 V-loads feed 32 PV-WMMAs.
// Online softmax with 16-lane shuffles; P staged through LDS for the
// C-layout -> A-layout transpose.
// ---------------------------------------------------------------------------
__global__ __launch_bounds__(32) void attn_kernel(
    const float* __restrict__ q, const float* __restrict__ k,
    const float* __restrict__ v, float* __restrict__ out)
{
    __shared__ float pstage[2 * 256];

    const int lane   = threadIdx.x & 31;
    const int lane16 = lane & 15;
    const int laneHi = lane >> 4;

    const int wid = blockIdx.x;          // (b*H + h)*(T/32) + q32
    const int q32 = wid & 63;            // T/32 = 64
    const int bh  = wid >> 6;            // 0..63
    const int b   = bh >> 4;
    const int h   = bh & 15;
    const int q0  = q32 * 32;

    // Preload Q fragments for both query halves (A-operand, K = 64 dims / 4)
    const float* qp0 = q + ((long)bh * SEQ + q0 + lane16) * HD + 2 * laneHi;
    const float* qp1 = qp0 + 16 * HD;
    v2f aQ0[16], aQ1[16];
    #pragma unroll
    for (int j = 0; j < 16; ++j) {
        aQ0[j] = *(const v2f*)(qp0 + 4 * j);
        aQ1[j] = *(const v2f*)(qp1 + 4 * j);
    }

    float mrow[2][8], lrow[2][8];
    v8f O[2][4] = {};
    #pragma unroll
    for (int g = 0; g < 2; ++g)
        #pragma unroll
        for (int vv = 0; vv < 8; ++vv) { mrow[g][vv] = -__builtin_inff(); lrow[g][vv] = 0.f; }

    const float scale = 0.125f;              // 1/sqrt(64)
    const int ktmax = (q0 + 31) >> 4;        // inclusive last 16-key tile

    for (int kt = 0; kt <= ktmax; ++kt) {
        const int k0 = kt * 16;

        // ---- S[g] = Q[g] K^T : K fragments shared across both halves ----
        const float* kp = k + ((long)bh * SEQ + k0 + lane16) * HD + 2 * laneHi;
        v8f S[2] = {};
        #pragma unroll
        for (int j = 0; j < 16; ++j) {
            v2f bK = *(const v2f*)(kp + 4 * j);
            S[0] = WMMA_F32(aQ0[j], bK, S[0]);
            S[1] = WMMA_F32(aQ1[j], bK, S[1]);
        }

        // ---- scale + causal mask + online softmax per half ----
        #pragma unroll
        for (int g = 0; g < 2; ++g) {
            #pragma unroll
            for (int vv = 0; vv < 8; ++vv) {
                const int row = 16 * g + vv + 8 * laneHi;   // row within 32-tile
                float s = S[g][vv] * scale;
                if (k0 + lane16 > q0 + row) s = -__builtin_inff();
                S[g][vv] = s;
            }
            float alpha[8];
            #pragma unroll
            for (int vv = 0; vv < 8; ++vv) {
                float tmax = S[g][vv];
                #pragma unroll
                for (int off = 1; off < 16; off <<= 1)
                    tmax = fmaxf(tmax, __shfl_xor(tmax, off, 32));
                const float mnew = fmaxf(mrow[g][vv], tmax);
                const float p    = __expf(S[g][vv] - mnew);   // masked -> 0
                float psum = p;
                #pragma unroll
                for (int off = 1; off < 16; off <<= 1)
                    psum += __shfl_xor(psum, off, 32);
                const float a = __expf(mrow[g][vv] - mnew);   // first tile -> 0
                lrow[g][vv]  = lrow[g][vv] * a + psum;
                mrow[g][vv]  = mnew;
                alpha[vv]    = a;
                S[g][vv]     = p;                             // S now holds P
            }
            #pragma unroll
            for (int c = 0; c < 4; ++c)
                #pragma unroll
                for (int vv = 0; vv < 8; ++vv)
                    O[g][c][vv] *= alpha[vv];
        }

        // ---- transpose P (both halves): C-layout -> A-layout via LDS ----
        __syncthreads();
        #pragma unroll
        for (int g = 0; g < 2; ++g)
            #pragma unroll
            for (int vv = 0; vv < 8; ++vv)
                pstage[g * 256 + (vv + 8 * laneHi) * 16 + lane16] = S[g][vv];
        __syncthreads();
        v2f aP0[4], aP1[4];
        #pragma unroll
        for (int j = 0; j < 4; ++j) {
            aP0[j] = *(const v2f*)(&pstage[lane16 * 16 + 4 * j + 2 * laneHi]);
            aP1[j] = *(const v2f*)(&pstage[256 + lane16 * 16 + 4 * j + 2 * laneHi]);
        }

        // ---- O[g] += P[g] V : V fragments shared across both halves ----
        const float* vp = v + ((long)bh * SEQ + k0) * HD;
        #pragma unroll
        for (int c = 0; c < 4; ++c) {
            #pragma unroll
            for (int j = 0; j < 4; ++j) {
                const int kk = 4 * j + 2 * laneHi;
                const float* vbp = vp + (long)kk * HD + c * 16 + lane16;
                v2f bV;
                bV.x = vbp[0];
                bV.y = vbp[HD];
                O[0][c] = WMMA_F32(aP0[j], bV, O[0][c]);
                O[1][c] = WMMA_F32(aP1[j], bV, O[1][c]);
            }
        }
    }

    // ---- epilogue: normalize and store [B, T, H, hd] ----
    #pragma unroll
    for (int g = 0; g < 2; ++g) {
        #pragma unroll
        for (int c = 0; c < 4; ++c) {
            #pragma unroll
            for (int vv = 0; vv < 8; ++vv) {
                const int t = q0 + 16 * g + vv + 8 * laneHi;
                const int d = c * 16 + lane16;
                out[(((long)b * SEQ + t) * NHEAD + h) * HD + d] = O[g][c][vv] / lrow[g][vv];
            }
        }
    }
}

extern "C" void kernel_launch(void* const* d_in, const int* in_sizes, int n_in,
                              void* d_out, int out_size, void* d_ws, size_t ws_size,
                              hipStream_t stream) {
    const float* x    = (const float*)d_in[0];
    const float* Wqkv = (const float*)d_in[1];
    const float* bqkv = (const float*)d_in[2];
    float* out = (float*)d_out;

    float* ws = (float*)d_ws;
    const long QS = (long)BATCH * NHEAD * SEQ * HD;   // 8,388,608 elems = 32 MB
    float* qb = ws;
    float* kb = ws + QS;
    float* vb = ws + 2 * QS;

    // 256 M-tiles * 48 N-tiles = 12288 waves, 8 waves/block (same N-tile per block)
    qkv_gemm_kernel<<<1536, 256, 0, stream>>>(x, Wqkv, bqkv, qb, kb, vb);
    // 4 * 16 * 64 query tiles of 32, one wave each
    attn_kernel<<<BATCH * NHEAD * (SEQ / 32), 32, 0, stream>>>(qb, kb, vb, out);
}